// BottleneckBase_25580825215706
// MI455X (gfx1250) — compile-verified
//
#include <hip/hip_runtime.h>

typedef __attribute__((ext_vector_type(2))) float v2f;
typedef __attribute__((ext_vector_type(8))) float v8f;

#define CIN  256
#define CMID 64
#define NOFF 27

__device__ __forceinline__ v8f wmma4(v2f a, v2f b, v8f c) {
  // D = A(16x4) * B(4x16) + C(16x16), fp32, wave32
  return __builtin_amdgcn_wmma_f32_16x16x4_f32(
      /*neg_a=*/false, a, /*neg_b=*/false, b,
      /*c_mod=*/(short)0, c, /*reuse_a=*/false, /*reuse_b=*/false);
}

// ---- CDNA5 async global->LDS copy (ASYNCcnt path, bypasses VGPRs) ----------
__device__ __forceinline__ unsigned lds_lo32(const void* p) {
  // generic (flat) shared pointer: addr[31:0] is the LDS offset (ISA 10.2)
  return (unsigned)(uintptr_t)p;
}
__device__ __forceinline__ void async_copy_b128(unsigned lds_byte_off,
                                                const void* gaddr) {
  asm volatile("global_load_async_to_lds_b128 %0, %1, off"
               :: "v"(lds_byte_off), "v"(gaddr)
               : "memory");
}
__device__ __forceinline__ void async_wait0() {
  asm volatile("s_wait_asynccnt 0" ::: "memory");
}

// ---------------------------------------------------------------- utilities
__global__ void k_zero(float* __restrict__ p, size_t total) {
  size_t i = (size_t)blockIdx.x * blockDim.x + threadIdx.x;
  size_t stride = (size_t)gridDim.x * blockDim.x;
  for (; i < total; i += stride) p[i] = 0.0f;
}

__global__ void k_finalize(const float* __restrict__ sum, const float* __restrict__ sq,
                           float* __restrict__ mu, float* __restrict__ rs,
                           int C, float invN) {
  int i = blockIdx.x * blockDim.x + threadIdx.x;
  if (i < C) {
    float m = sum[i] * invN;
    float v = sq[i] * invN - m * m;
    mu[i] = m;
    rs[i] = rsqrtf(v + 1e-5f);
  }
}

// in-place y = relu(gamma*(y-mu)*rstd + beta), C = cmask+1 channels (row-major)
__global__ void k_bnrelu(float* __restrict__ a, const float* __restrict__ g,
                         const float* __restrict__ b, const float* __restrict__ mu,
                         const float* __restrict__ rs, size_t total, int cmask) {
  size_t i = (size_t)blockIdx.x * blockDim.x + threadIdx.x;
  size_t stride = (size_t)gridDim.x * blockDim.x;
  for (; i < total; i += stride) {
    int c = (int)(i & (size_t)cmask);
    float v = (a[i] - mu[c]) * rs[c] * g[c] + b[c];
    a[i] = v > 0.0f ? v : 0.0f;
  }
}

// column sums/sumsq for a [n,64] row-major matrix
__global__ __launch_bounds__(256) void k_colstats64(const float* __restrict__ a, int n,
                                                    float* __restrict__ sum,
                                                    float* __restrict__ sq) {
  __shared__ float ls[64], lss[64];
  int tid = threadIdx.x;
  if (tid < 64) { ls[tid] = 0.0f; lss[tid] = 0.0f; }
  __syncthreads();
  int col = tid & 63;  // stride is a multiple of 64 -> column invariant per thread
  float s = 0.0f, ss = 0.0f;
  size_t total = (size_t)n * 64;
  size_t stride = (size_t)gridDim.x * blockDim.x;
  for (size_t i = (size_t)blockIdx.x * blockDim.x + tid; i < total; i += stride) {
    float v = a[i];
    s += v; ss += v * v;
  }
  atomicAdd(&ls[col], s);
  atomicAdd(&lss[col], ss);
  __syncthreads();
  if (tid < 64) { atomicAdd(&sum[tid], ls[tid]); atomicAdd(&sq[tid], lss[tid]); }
}

// ------------------- conv1: y1 = x @ W1, + BN stats (32 rows/block, 2 tiles)
__global__ __launch_bounds__(128) void k_conv1(const float* __restrict__ x,
                                               const float* __restrict__ W1,
                                               float* __restrict__ y1,
                                               float* __restrict__ sum1,
                                               float* __restrict__ sq1) {
  __shared__ __align__(16) float xs[32 * CIN];  // 32 KB tile of x
  int row0 = blockIdx.x * 32;
  int tid = threadIdx.x;
  {  // async-stage 32 contiguous rows (32 KB = 2048 x b128)
    unsigned lb = lds_lo32(xs);
    const char* g = (const char*)(x + (size_t)row0 * CIN);
    for (int i = tid; i < 2048; i += 128)
      async_copy_b128(lb + (unsigned)i * 16u, g + (size_t)i * 16);
    async_wait0();
  }
  __syncthreads();

  int wave = tid >> 5, lane = tid & 31, half = lane >> 4, l = lane & 15;
  int n0 = wave * 16;
  v8f c0 = {}, c1 = {};
  for (int k = 0; k < CIN; k += 4) {
    int ka = k + 2 * half;
    v2f a0 = *(const v2f*)&xs[l * CIN + ka];
    v2f a1 = *(const v2f*)&xs[(16 + l) * CIN + ka];
    v2f bb;  // B fragment shared by both row-tiles
    bb[0] = W1[(size_t)ka * CMID + n0 + l];
    bb[1] = W1[(size_t)(ka + 1) * CMID + n0 + l];
    c0 = wmma4(a0, bb, c0);
    c1 = wmma4(a1, bb, c1);
  }
  float s = 0.0f, ss = 0.0f;
#pragma unroll
  for (int j = 0; j < 8; ++j) {
    int r = row0 + j + half * 8;
    float v = c0[j];
    y1[(size_t)r * CMID + n0 + l] = v;
    s += v; ss += v * v;
    v = c1[j];
    y1[(size_t)(r + 16) * CMID + n0 + l] = v;
    s += v; ss += v * v;
  }
  s += __shfl_xor(s, 16, 32);
  ss += __shfl_xor(ss, 16, 32);
  if (half == 0) {
    atomicAdd(&sum1[n0 + l], s);
    atomicAdd(&sq1[n0 + l], ss);
  }
}

// -- conv2: async-gather h -> WMMA vs W2[k] -> scatter-add (32 msgs/block)
__global__ __launch_bounds__(128) void k_conv2(const float* __restrict__ h,
                                               const float* __restrict__ W2,
                                               const int* __restrict__ in_map,
                                               const int* __restrict__ out_map,
                                               float* __restrict__ out2, int m) {
  __shared__ __align__(16) float hs[32 * CMID];  // gathered 32x64 tile (8 KB)
  __shared__ int iidx[32], oidx[32];
  int k = blockIdx.y;
  int m0 = blockIdx.x * 32;
  int tid = threadIdx.x;
  if (tid < 32) {
    iidx[tid] = in_map[(size_t)k * m + m0 + tid];
    oidx[tid] = out_map[(size_t)k * m + m0 + tid];
  }
  __syncthreads();
  {  // async gather: 32 rows x 16 b128 (256 B/row)
    unsigned lb = lds_lo32(hs);
    for (int i = tid; i < 512; i += 128) {
      int r = i >> 4, q = i & 15;
      async_copy_b128(lb + (unsigned)i * 16u,
                      (const char*)(h + (size_t)iidx[r] * CMID) + q * 16);
    }
    async_wait0();
  }
  __syncthreads();

  int wave = tid >> 5, lane = tid & 31, half = lane >> 4, l = lane & 15;
  int n0 = wave * 16;
  const float* Wk = W2 + (size_t)k * CMID * CMID;
  v8f c0 = {}, c1 = {};
#pragma unroll 4
  for (int kk = 0; kk < CMID; kk += 4) {
    int ka = kk + 2 * half;
    v2f a0 = *(const v2f*)&hs[l * CMID + ka];
    v2f a1 = *(const v2f*)&hs[(16 + l) * CMID + ka];
    v2f bb;
    bb[0] = Wk[ka * CMID + n0 + l];
    bb[1] = Wk[(ka + 1) * CMID + n0 + l];
    c0 = wmma4(a0, bb, c0);
    c1 = wmma4(a1, bb, c1);
  }
#pragma unroll
  for (int j = 0; j < 8; ++j) {
    int r0 = oidx[j + half * 8];
    int r1 = oidx[16 + j + half * 8];
    atomicAdd(&out2[(size_t)r0 * CMID + n0 + l], c0[j]);
    atomicAdd(&out2[(size_t)r1 * CMID + n0 + l], c1[j]);
  }
}

// ------- conv3 shared tile-GEMM: [32,64] x [64,256] column-group (2 tiles)
__device__ __forceinline__ void conv3_tiles(const float* __restrict__ as,
                                            const float* __restrict__ W3,
                                            int n0, int half, int l,
                                            v8f& c0, v8f& c1) {
#pragma unroll 4
  for (int k = 0; k < CMID; k += 4) {
    int ka = k + 2 * half;
    v2f a0 = *(const v2f*)&as[l * CMID + ka];
    v2f a1 = *(const v2f*)&as[(16 + l) * CMID + ka];
    v2f bb;
    bb[0] = W3[(size_t)ka * CIN + n0 + l];
    bb[1] = W3[(size_t)(ka + 1) * CIN + n0 + l];
    c0 = wmma4(a0, bb, c0);
    c1 = wmma4(a1, bb, c1);
  }
}

__device__ __forceinline__ void conv3_stage(float* as, const float* a64,
                                            int row0, int tid) {
  unsigned lb = lds_lo32(as);
  const char* g = (const char*)(a64 + (size_t)row0 * CMID);
  for (int i = tid; i < 512; i += 128)  // 32 rows x 64 f32 = 8 KB
    async_copy_b128(lb + (unsigned)i * 16u, g + (size_t)i * 16);
  async_wait0();
}

__global__ __launch_bounds__(128) void k_conv3_stats(const float* __restrict__ a64,
                                                     const float* __restrict__ W3,
                                                     float* __restrict__ sum3,
                                                     float* __restrict__ sq3) {
  __shared__ __align__(16) float as[32 * CMID];
  int row0 = blockIdx.x * 32;
  int tid = threadIdx.x;
  conv3_stage(as, a64, row0, tid);
  __syncthreads();
  int wave = tid >> 5, lane = tid & 31, half = lane >> 4, l = lane & 15;
  int n0 = blockIdx.y * 64 + wave * 16;
  v8f c0 = {}, c1 = {};
  conv3_tiles(as, W3, n0, half, l, c0, c1);
  float s = 0.0f, ss = 0.0f;
#pragma unroll
  for (int j = 0; j < 8; ++j) {
    float v = c0[j]; s += v; ss += v * v;
    v = c1[j];       s += v; ss += v * v;
  }
  s += __shfl_xor(s, 16, 32);
  ss += __shfl_xor(ss, 16, 32);
  if (half == 0) {
    atomicAdd(&sum3[n0 + l], s);
    atomicAdd(&sq3[n0 + l], ss);
  }
}

__global__ __launch_bounds__(128) void k_conv3_final(const float* __restrict__ a64,
                                                     const float* __restrict__ W3,
                                                     const float* __restrict__ g3,
                                                     const float* __restrict__ b3,
                                                     const float* __restrict__ mu3,
                                                     const float* __restrict__ rs3,
                                                     const float* __restrict__ x,
                                                     float* __restrict__ out) {
  __shared__ __align__(16) float as[32 * CMID];
  int row0 = blockIdx.x * 32;
  int tid = threadIdx.x;
  conv3_stage(as, a64, row0, tid);
  __syncthreads();
  int wave = tid >> 5, lane = tid & 31, half = lane >> 4, l = lane & 15;
  int n0 = blockIdx.y * 64 + wave * 16;
  v8f c0 = {}, c1 = {};
  conv3_tiles(as, W3, n0, half, l, c0, c1);
  int col = n0 + l;
  float mm = mu3[col], rr = rs3[col], gg = g3[col], bb = b3[col];
#pragma unroll
  for (int j = 0; j < 8; ++j) {
    int r = row0 + j + half * 8;
    float v = (c0[j] - mm) * rr * gg + bb + x[(size_t)r * CIN + col];
    out[(size_t)r * CIN + col] = v > 0.0f ? v : 0.0f;
    v = (c1[j] - mm) * rr * gg + bb + x[(size_t)(r + 16) * CIN + col];
    out[(size_t)(r + 16) * CIN + col] = v > 0.0f ? v : 0.0f;
  }
}

// ---------------------------------------------------------------- launcher
extern "C" void kernel_launch(void* const* d_in, const int* in_sizes, int n_in,
                              void* d_out, int out_size, void* d_ws, size_t ws_size,
                              hipStream_t stream) {
  const float* x  = (const float*)d_in[0];
  const float* W1 = (const float*)d_in[1];
  const float* g1 = (const float*)d_in[2];
  const float* b1 = (const float*)d_in[3];
  const float* W2 = (const float*)d_in[4];
  const float* g2 = (const float*)d_in[5];
  const float* b2 = (const float*)d_in[6];
  const float* W3 = (const float*)d_in[7];
  const float* g3 = (const float*)d_in[8];
  const float* b3 = (const float*)d_in[9];
  const int* in_map  = (const int*)d_in[10];
  const int* out_map = (const int*)d_in[11];
  int n = in_sizes[0] / CIN;       // 300000
  int m = in_sizes[10] / NOFF;     // 100000
  float* out = (float*)d_out;

  // workspace layout (floats)
  float* ws   = (float*)d_ws;
  float* y1   = ws;                          // [n,64]  (becomes h in-place)
  float* out2 = ws + (size_t)n * CMID;       // [n,64]  (becomes bn2 output in-place)
  float* st   = out2 + (size_t)n * CMID;     // stats
  float *sum1 = st,        *sq1 = st + 64,   *mu1 = st + 128,  *rs1 = st + 192;
  float *sum2 = st + 256,  *sq2 = st + 320,  *mu2 = st + 384,  *rs2 = st + 448;
  float *sum3 = st + 512,  *sq3 = st + 768,  *mu3 = st + 1024, *rs3 = st + 1280;
  const float invN = 1.0f / (float)n;

  // 1) zero out2 + entire stats block (contiguous after out2)
  k_zero<<<2048, 256, 0, stream>>>(out2, (size_t)n * CMID + 1536);

  // 2) conv1 + stats1
  k_conv1<<<n / 32, 128, 0, stream>>>(x, W1, y1, sum1, sq1);
  k_finalize<<<1, 64, 0, stream>>>(sum1, sq1, mu1, rs1, CMID, invN);
  k_bnrelu<<<2048, 256, 0, stream>>>(y1, g1, b1, mu1, rs1, (size_t)n * CMID, CMID - 1);

  // 3) conv2 async-gather -> WMMA -> scatter-add
  k_conv2<<<dim3(m / 32, NOFF), 128, 0, stream>>>(y1, W2, in_map, out_map, out2, m);

  // 4) BN2
  k_colstats64<<<1024, 256, 0, stream>>>(out2, n, sum2, sq2);
  k_finalize<<<1, 64, 0, stream>>>(sum2, sq2, mu2, rs2, CMID, invN);
  k_bnrelu<<<2048, 256, 0, stream>>>(out2, g2, b2, mu2, rs2, (size_t)n * CMID, CMID - 1);

  // 5) conv3 stats pass (y3 discarded; recompute beats a 307 MB round-trip)
  k_conv3_stats<<<dim3(n / 32, CIN / 64), 128, 0, stream>>>(out2, W3, sum3, sq3);
  k_finalize<<<1, 256, 0, stream>>>(sum3, sq3, mu3, rs3, CIN, invN);

  // 6) conv3 final: recompute + BN3 + residual + relu -> d_out
  k_conv3_final<<<dim3(n / 32, CIN / 64), 128, 0, stream>>>(out2, W3, g3, b3, mu3, rs3, x, out);
}